// GraphPositionalEncoding_11269994184783
// MI455X (gfx1250) — compile-verified
//
#include <hip/hip_runtime.h>
#include <stdint.h>

// GraphPositionalEncoding: out[n,h,l,s] = QK[n,h,l,s] + table[pos[n,l,s]][h]
// table = emb_table with row 0 zeroed (padding_idx=0).
// N=8 H=32 L=512 S=512, MAX_SPATIAL=100.
// Pure HBM-streaming op (~520 MB moved, ~22us at 23.3 TB/s). Strategy:
//  - b128 non-temporal loads/stores for the 512MB QK/out streams
//  - emb_table staged in LDS with 33-float padded rows (bank-conflict-free gather)
//  - async global->LDS staging via gfx1250 builtin (int* params per hipcc diagnostic)
//  - grid-stride (2048 blocks) to amortize table staging

#define N_    8
#define H_    32
#define L_    512
#define S_    512
#define MAXP  100
#define LSV   (L_ * S_ / 4)            // 65536 vec4 per (n,h) slice
#define NV    (N_ * H_ * L_ * S_ / 4)  // 16,777,216 vec4 elements total
#define PADW  33                       // padded LDS row width (spreads banks: 33 invertible mod 64)
#define LUTN  (MAXP * PADW)            // 3300 floats = 13.2 KB LDS

typedef float v4f __attribute__((ext_vector_type(4)));
typedef int   v4i __attribute__((ext_vector_type(4)));

typedef __attribute__((address_space(1))) int  gint;
typedef __attribute__((address_space(3))) int  lint;

__global__ __launch_bounds__(256) void gpe_bias_add_kernel(
    const v4f* __restrict__ qk,
    const v4i* __restrict__ pos4,
    const float* __restrict__ emb,
    v4f* __restrict__ out)
{
    __shared__ float lut[LUTN];
    const int tid = threadIdx.x;

    // ---- Stage emb_table (100x32) into LDS, padded to 33 floats/row ----
#if __has_builtin(__builtin_amdgcn_global_load_async_to_lds_b32)
    // CDNA5 async DMA path: per-lane 4B global->LDS copies, tracked by ASYNCcnt.
    for (int j = tid; j < MAXP * H_; j += 256) {
        const int row = j >> 5;          // /32
        const int col = j & 31;
        // addrspace(3) pointer value == LDS byte offset (low 32 bits of flat addr)
        __builtin_amdgcn_global_load_async_to_lds_b32(
            (gint*)(uintptr_t)(emb + j),
            (lint*)(uint32_t)(uintptr_t)&lut[row * PADW + col],
            0, 0);
    }
    asm volatile("s_wait_asynccnt 0" ::: "memory");
    // padding_idx=0: overwrite row 0 with zeros after the DMA lands
    for (int j = tid; j < H_; j += 256) lut[j] = 0.0f;
#else
    // Fallback: plain ds stores
    for (int j = tid; j < MAXP * H_; j += 256) {
        const int row = j >> 5;
        const int col = j & 31;
        lut[row * PADW + col] = (row == 0) ? 0.0f : emb[j];
    }
#endif
    __syncthreads();

    // ---- Streaming gather+add over vec4 elements ----
    const int gstride = (int)(gridDim.x * 256u);
    for (int i = (int)(blockIdx.x * 256u) + tid; i < NV; i += gstride) {
        const int h   = (i >> 16) & (H_ - 1);   // LSV = 2^16
        const int n   = i >> 21;                // H_*LSV = 2^21
        const int rem = i & (LSV - 1);

        const v4i p = pos4[(n << 16) | rem];          // temporal: reused across 32 h-slices via L2
        const v4f q = __builtin_nontemporal_load(&qk[i]);  // NT: don't pollute L2

        v4f r;
        r.x = q.x + lut[p.x * PADW + h];
        r.y = q.y + lut[p.y * PADW + h];
        r.z = q.z + lut[p.z * PADW + h];
        r.w = q.w + lut[p.w * PADW + h];

        __builtin_nontemporal_store(r, &out[i]);      // NT store
    }
}

extern "C" void kernel_launch(void* const* d_in, const int* in_sizes, int n_in,
                              void* d_out, int out_size, void* d_ws, size_t ws_size,
                              hipStream_t stream) {
    const v4f*   qk   = (const v4f*)d_in[0];   // [8,32,512,512] f32
    const v4i*   pos4 = (const v4i*)d_in[1];   // [8,512,512] i32, viewed as vec4
    const float* emb  = (const float*)d_in[2]; // [100,32] f32
    v4f*         out  = (v4f*)d_out;           // [8,32,512,512] f32

    // 2048 blocks x 256 threads (8 wave32 per block); each thread does 32 vec4 iters.
    gpe_bias_add_kernel<<<dim3(2048), dim3(256), 0, stream>>>(qk, pos4, emb, out);
}